// arch51_71768903516181
// MI455X (gfx1250) — compile-verified
//
#include <hip/hip_runtime.h>
#include <math.h>

// ---------------------------------------------------------------------------
// MI455X (gfx1250) implementation: spectral recombine -> 2D IFFT via radix-16
// WMMA (V_WMMA_F32_16X16X4_F32) -> conv chain -> FC head.
// Twiddles use hardware sin/cos (__sincosf): angles are pre-reduced to
// [0, 2pi) via integer masks, so no libm argument-reduction slow path.
// ---------------------------------------------------------------------------

typedef __attribute__((ext_vector_type(2))) float v2f;
typedef __attribute__((ext_vector_type(8))) float v8f;

#define PI_F 3.14159265358979323846f
#define FFT_N 4096

__device__ inline v8f wmma4(v2f a, v2f b, v8f c) {
  // D = A(16x4,f32) x B(4x16,f32) + C(16x16,f32)
  return __builtin_amdgcn_wmma_f32_16x16x4_f32(false, a, false, b, (short)0, c,
                                               false, false);
}

// Per-lane A-operand fragments of the inverse DFT-16 matrix F[m][k]=e^{+2pi i mk/16}.
// Layout (ISA 7.12.2, 32-bit A 16x4): lane<16 holds (K=kk+0,kk+1), lane>=16 (K=kk+2,kk+3).
__device__ inline void dft16_a_frags(int lane, v2f* aR, v2f* aI, v2f* aIn) {
  int m = lane & 15;
  int kb = (lane >> 4) << 1;
#pragma unroll
  for (int c = 0; c < 4; ++c) {
    int k0 = c * 4 + kb;
    float s0, c0, s1, c1;
    __sincosf((2.0f * PI_F / 16.0f) * (float)((m * k0) & 15), &s0, &c0);
    __sincosf((2.0f * PI_F / 16.0f) * (float)((m * (k0 + 1)) & 15), &s1, &c1);
    aR[c].x = c0; aR[c].y = c1;
    aI[c].x = s0; aI[c].y = s1;
    aIn[c].x = -s0; aIn[c].y = -s1;
  }
}

// One complex 16-point DFT tile (16 DFTs batched over N=16 columns) in LDS,
// in place, with twiddle angle = twFac * ((m * (jbase+n)) & mask).
__device__ inline void dft16_tile_lds(float* sRe, float* sIm, int base,
                                      int strideK, int strideN, int strideM,
                                      float twFac, int jbase, int mask,
                                      const v2f* aR, const v2f* aI,
                                      const v2f* aIn, int lane) {
  const int n = lane & 15;
  const int kb = (lane >> 4) << 1;
  const int hi = lane >> 4;
  v8f accR = {0.f, 0.f, 0.f, 0.f, 0.f, 0.f, 0.f, 0.f};
  v8f accI = {0.f, 0.f, 0.f, 0.f, 0.f, 0.f, 0.f, 0.f};
#pragma unroll
  for (int c = 0; c < 4; ++c) {
    int k0 = c * 4 + kb;
    int i0 = base + k0 * strideK + n * strideN;
    int i1 = i0 + strideK;
    v2f bR, bI;
    bR.x = sRe[i0]; bR.y = sRe[i1];
    bI.x = sIm[i0]; bI.y = sIm[i1];
    accR = wmma4(aR[c], bR, accR);   // + Fr*Xr
    accR = wmma4(aIn[c], bI, accR);  // - Fi*Xi
    accI = wmma4(aR[c], bI, accI);   // + Fr*Xi
    accI = wmma4(aI[c], bR, accI);   // + Fi*Xr
  }
#pragma unroll
  for (int i = 0; i < 8; ++i) {
    int mi = i + 8 * hi;
    float ang = twFac * (float)((mi * (jbase + n)) & mask);  // in [0, 2pi)
    float sn, cs;
    __sincosf(ang, &sn, &cs);
    float yr = accR[i] * cs - accI[i] * sn;
    float yi = accR[i] * sn + accI[i] * cs;
    int w = base + mi * strideM + n * strideN;
    sRe[w] = yr;
    sIm[w] = yi;
  }
}

// Global-memory column-FFT tile: rows r = rk_base + k*rk_stride, 16 consecutive
// image columns starting at col0; in place; uniform twiddle index jval.
__device__ inline void dft16_tile_global(float* gRe, float* gIm, int rk_base,
                                         int rk_stride, int col0, float twFac,
                                         int jval, int mask, const v2f* aR,
                                         const v2f* aI, const v2f* aIn,
                                         int lane) {
  const int n = lane & 15;
  const int kb = (lane >> 4) << 1;
  const int hi = lane >> 4;
  v8f accR = {0.f, 0.f, 0.f, 0.f, 0.f, 0.f, 0.f, 0.f};
  v8f accI = {0.f, 0.f, 0.f, 0.f, 0.f, 0.f, 0.f, 0.f};
#pragma unroll
  for (int c = 0; c < 4; ++c) {
    int k0 = c * 4 + kb;
    size_t r0 = (size_t)(rk_base + k0 * rk_stride) * FFT_N + col0 + n;
    size_t r1 = (size_t)(rk_base + (k0 + 1) * rk_stride) * FFT_N + col0 + n;
    v2f bR, bI;
    bR.x = gRe[r0]; bR.y = gRe[r1];
    bI.x = gIm[r0]; bI.y = gIm[r1];
    accR = wmma4(aR[c], bR, accR);
    accR = wmma4(aIn[c], bI, accR);
    accI = wmma4(aR[c], bI, accI);
    accI = wmma4(aI[c], bR, accI);
  }
#pragma unroll
  for (int i = 0; i < 8; ++i) {
    int mi = i + 8 * hi;
    float ang = twFac * (float)((mi * jval) & mask);  // in [0, 2pi)
    float sn, cs;
    __sincosf(ang, &sn, &cs);
    float yr = accR[i] * cs - accI[i] * sn;
    float yi = accR[i] * sn + accI[i] * cs;
    size_t w = (size_t)(rk_base + mi * rk_stride) * FFT_N + col0 + n;
    gRe[w] = yr;
    gIm[w] = yi;
  }
}

// -------------------- Kernel 1: spectral recombination (float4) -------------
__global__ __launch_bounds__(256) void prep_kernel(
    const float4* __restrict__ xr, const float4* __restrict__ xi,
    const float4* __restrict__ mk, const float4* __restrict__ pk,
    float4* __restrict__ oRe, float4* __restrict__ oIm) {
  int idx = blockIdx.x * 256 + threadIdx.x;
  float4 a4 = xr[idx], b4 = xi[idx], m4 = mk[idx], p4 = pk[idx];
  float4 r4, i4;
  const float* a = &a4.x;
  const float* b = &b4.x;
  const float* m = &m4.x;
  const float* p = &p4.x;
  float* rr = &r4.x;
  float* ii = &i4.x;
#pragma unroll
  for (int c = 0; c < 4; ++c) {
    // fold the 1/N^2 inverse-FFT normalization in here
    float mag = sqrtf(a[c] * a[c] + b[c] * b[c]) * m[c] * (1.0f / 16777216.0f);
    float ph = atan2f(b[c], a[c]) * p[c];
    float s, cs;
    __sincosf(ph, &s, &cs);
    rr[c] = mag * cs;
    ii[c] = mag * s;
  }
  oRe[idx] = r4;
  oIm[idx] = i4;
}

// -------------------- Kernel 2: row IFFT (LDS, 3 radix-16 stages) -----------
__global__ __launch_bounds__(256) void fft_rows_kernel(float* __restrict__ gRe,
                                                       float* __restrict__ gIm) {
  __shared__ float sRe[FFT_N];
  __shared__ float sIm[FFT_N];
  const int row = blockIdx.x;
  const int tid = threadIdx.x;
  const int lane = tid & 31;
  const int wave = tid >> 5;
  const size_t rb = (size_t)row * FFT_N;

  {  // vectorized load: 4096 floats/plane as 1024 float4
    const float4* gRe4 = reinterpret_cast<const float4*>(gRe + rb);
    const float4* gIm4 = reinterpret_cast<const float4*>(gIm + rb);
    for (int v = tid; v < FFT_N / 4; v += 256) {
      float4 r = gRe4[v], m = gIm4[v];
      sRe[v * 4 + 0] = r.x; sRe[v * 4 + 1] = r.y;
      sRe[v * 4 + 2] = r.z; sRe[v * 4 + 3] = r.w;
      sIm[v * 4 + 0] = m.x; sIm[v * 4 + 1] = m.y;
      sIm[v * 4 + 2] = m.z; sIm[v * 4 + 3] = m.w;
    }
  }
  __syncthreads();

  v2f aR[4], aI[4], aIn[4];
  dft16_a_frags(lane, aR, aI, aIn);

  // stage 0: stride-256 DFTs, twiddle W4096^(m*j), j = t*16+n
  for (int t = wave; t < 16; t += 8)
    dft16_tile_lds(sRe, sIm, t * 16, 256, 1, 256,
                   2.0f * PI_F / 4096.0f, t * 16, 4095, aR, aI, aIn, lane);
  __syncthreads();
  // stage 1: stride-16 DFTs inside each 256-block, twiddle W256^(m*j2)
  for (int q = wave; q < 16; q += 8)
    dft16_tile_lds(sRe, sIm, q * 256, 16, 1, 16,
                   2.0f * PI_F / 256.0f, 0, 255, aR, aI, aIn, lane);
  __syncthreads();
  // stage 2: contiguous 16-point DFTs, no twiddle (mask 0 -> angle 0)
  for (int t = wave; t < 16; t += 8)
    dft16_tile_lds(sRe, sIm, t * 256, 1, 16, 1, 0.0f, 0, 0, aR, aI, aIn, lane);
  __syncthreads();

  {  // write back with radix-16 digit reversal; 4 consecutive outputs come
     // from LDS at stride 256, so global stores vectorize to b128
    float4* gRe4 = reinterpret_cast<float4*>(gRe + rb);
    float4* gIm4 = reinterpret_cast<float4*>(gIm + rb);
    for (int v = tid; v < FFT_N / 4; v += 256) {
      int f = v * 4;
      int p = (f & 15) * 256 + (f & 240) + (f >> 8);
      float4 r, m;
      r.x = sRe[p];       r.y = sRe[p + 256];
      r.z = sRe[p + 512]; r.w = sRe[p + 768];
      m.x = sIm[p];       m.y = sIm[p + 256];
      m.z = sIm[p + 512]; m.w = sIm[p + 768];
      gRe4[v] = r;
      gIm4[v] = m;
    }
  }
}

// -------------------- Kernels 3-5: column IFFT, 3 global passes -------------
__global__ __launch_bounds__(256) void fft_cols_s0(float* __restrict__ gRe,
                                                   float* __restrict__ gIm) {
  int lane = threadIdx.x & 31, wave = threadIdx.x >> 5;
  int tile = blockIdx.x * 8 + wave;
  int ct = tile & 255, jIdx = tile >> 8;
  v2f aR[4], aI[4], aIn[4];
  dft16_a_frags(lane, aR, aI, aIn);
  dft16_tile_global(gRe, gIm, jIdx, 256, ct * 16,
                    2.0f * PI_F / 4096.0f, jIdx, 4095, aR, aI, aIn, lane);
}

__global__ __launch_bounds__(256) void fft_cols_s1(float* __restrict__ gRe,
                                                   float* __restrict__ gIm) {
  int lane = threadIdx.x & 31, wave = threadIdx.x >> 5;
  int tile = blockIdx.x * 8 + wave;
  int ct = tile & 255, qj = tile >> 8;
  int q = qj >> 4, j2 = qj & 15;
  v2f aR[4], aI[4], aIn[4];
  dft16_a_frags(lane, aR, aI, aIn);
  dft16_tile_global(gRe, gIm, q * 256 + j2, 16, ct * 16,
                    2.0f * PI_F / 256.0f, j2, 255, aR, aI, aIn, lane);
}

// Final column stage: only the real part is needed downstream. Applies the
// digit reversal on the row index and the (-1)^(r+c) factor that implements
// ifftshift on even N.
__global__ __launch_bounds__(256) void fft_cols_s2(
    const float* __restrict__ gRe, const float* __restrict__ gIm,
    float* __restrict__ outReal) {
  int lane = threadIdx.x & 31, wave = threadIdx.x >> 5;
  int tile = blockIdx.x * 8 + wave;
  int ct = tile & 255, g = tile >> 8;
  int col0 = ct * 16;
  v2f aR[4], aI[4], aIn[4];
  dft16_a_frags(lane, aR, aI, aIn);

  const int n = lane & 15;
  const int kb = (lane >> 4) << 1;
  const int hi = lane >> 4;
  v8f accR = {0.f, 0.f, 0.f, 0.f, 0.f, 0.f, 0.f, 0.f};
#pragma unroll
  for (int c = 0; c < 4; ++c) {
    int k0 = c * 4 + kb;
    size_t r0 = (size_t)(g * 16 + k0) * FFT_N + col0 + n;
    size_t r1 = r0 + FFT_N;
    v2f bR, bI;
    bR.x = gRe[r0]; bR.y = gRe[r1];
    bI.x = gIm[r0]; bI.y = gIm[r1];
    accR = wmma4(aR[c], bR, accR);   // Re = Fr*Xr - Fi*Xi
    accR = wmma4(aIn[c], bI, accR);
  }
  int ad = g >> 4, bd = g & 15;
#pragma unroll
  for (int i = 0; i < 8; ++i) {
    int mi = i + 8 * hi;
    int f = mi * 256 + bd * 16 + ad;  // digit-reversed output row
    int col = col0 + n;
    float sgn = ((f + col) & 1) ? -1.0f : 1.0f;
    outReal[(size_t)f * FFT_N + col] = sgn * accR[i];
  }
}

// -------------------- Kernel 6: conv + BN + PReLU (x5) ----------------------
__global__ __launch_bounds__(256) void conv_bn_prelu(
    const float* __restrict__ in, int inN, float* __restrict__ out, int outN,
    const float* __restrict__ w, const float* __restrict__ b,
    const float* __restrict__ gamma, const float* __restrict__ beta,
    const float* __restrict__ mean, const float* __restrict__ var,
    const float* __restrict__ pa, int layer) {
  int idx = blockIdx.x * 256 + threadIdx.x;
  if (idx >= outN * outN) return;
  int oy = idx / outN, ox = idx - oy * outN;
  const float* wk = w + layer * 9;
  int iy = oy * 3, ix = ox * 3;
  float s = 0.0f;
#pragma unroll
  for (int ky = 0; ky < 3; ++ky)
#pragma unroll
    for (int kx = 0; kx < 3; ++kx)
      s += in[(size_t)(iy + ky) * inN + (ix + kx)] * wk[ky * 3 + kx];
  s += b[layer];
  float rs = rsqrtf(var[layer] + 1e-5f);
  float sc = gamma[layer] * rs;
  float sh = beta[layer] - gamma[layer] * mean[layer] * rs;
  s = s * sc + sh;
  out[idx] = (s >= 0.0f) ? s : pa[layer] * s;
}

// -------------------- Kernel 7: FC -> LN -> PReLU -> FC ---------------------
__global__ __launch_bounds__(128) void head_kernel(
    const float* __restrict__ x256, const float* __restrict__ w5,
    const float* __restrict__ b5, const float* __restrict__ lg,
    const float* __restrict__ lb, const float* __restrict__ pa,
    const float* __restrict__ w6, const float* __restrict__ b6,
    float* __restrict__ out) {
  __shared__ float xs[256];
  __shared__ float h[82];
  __shared__ float red[2];
  int tid = threadIdx.x;
  xs[tid] = x256[tid];
  xs[tid + 128] = x256[tid + 128];
  __syncthreads();
  if (tid < 82) {
    float s = b5[tid];
    for (int i = 0; i < 256; ++i) s += xs[i] * w5[i * 82 + tid];
    h[tid] = s;
  }
  __syncthreads();
  if (tid == 0) {
    float mu = 0.0f;
    for (int j = 0; j < 82; ++j) mu += h[j];
    mu *= (1.0f / 82.0f);
    float var = 0.0f;
    for (int j = 0; j < 82; ++j) { float d = h[j] - mu; var += d * d; }
    var *= (1.0f / 82.0f);
    red[0] = mu;
    red[1] = rsqrtf(var + 1e-5f);
  }
  __syncthreads();
  if (tid < 82) {
    float v = (h[tid] - red[0]) * red[1] * lg[tid] + lb[tid];
    h[tid] = (v >= 0.0f) ? v : pa[0] * v;
  }
  __syncthreads();
  if (tid < 5) {
    float s = b6[tid];
    for (int j = 0; j < 82; ++j) s += h[j] * w6[j * 5 + tid];
    out[tid] = s;
  }
}

// ---------------------------------------------------------------------------
extern "C" void kernel_launch(void* const* d_in, const int* in_sizes, int n_in,
                              void* d_out, int out_size, void* d_ws,
                              size_t ws_size, hipStream_t stream) {
  const float* x_real = (const float*)d_in[0];
  const float* x_imag = (const float*)d_in[1];
  const float* mag_k  = (const float*)d_in[2];
  const float* phs_k  = (const float*)d_in[3];
  const float* conv_w = (const float*)d_in[4];
  const float* conv_b = (const float*)d_in[5];
  const float* bn_g   = (const float*)d_in[6];
  const float* bn_b   = (const float*)d_in[7];
  const float* bn_m   = (const float*)d_in[8];
  const float* bn_v   = (const float*)d_in[9];
  const float* prelua = (const float*)d_in[10];
  const float* fc5_w  = (const float*)d_in[11];
  const float* fc5_b  = (const float*)d_in[12];
  const float* ln_g   = (const float*)d_in[13];
  const float* ln_b   = (const float*)d_in[14];
  const float* pre5   = (const float*)d_in[15];
  const float* fc6_w  = (const float*)d_in[16];
  const float* fc6_b  = (const float*)d_in[17];
  float* out = (float*)d_out;

  const size_t NN = (size_t)FFT_N * FFT_N;
  float* P0 = (float*)d_ws;        // real plane (in-place FFT)
  float* P1 = P0 + NN;             // imag plane
  float* P2 = P1 + NN;             // final real image
  float* C1 = P2 + NN;             // 1365^2
  float* C2 = C1 + 1365 * 1365;    // 455^2
  float* C3 = C2 + 455 * 455;      // 151^2
  float* C4 = C3 + 151 * 151;      // 50^2
  float* C5 = C4 + 50 * 50;        // 16^2

  prep_kernel<<<(int)(NN / 4 / 256), 256, 0, stream>>>(
      (const float4*)x_real, (const float4*)x_imag, (const float4*)mag_k,
      (const float4*)phs_k, (float4*)P0, (float4*)P1);
  fft_rows_kernel<<<FFT_N, 256, 0, stream>>>(P0, P1);
  fft_cols_s0<<<8192, 256, 0, stream>>>(P0, P1);
  fft_cols_s1<<<8192, 256, 0, stream>>>(P0, P1);
  fft_cols_s2<<<8192, 256, 0, stream>>>(P0, P1, P2);

  const int dims[6] = {4096, 1365, 455, 151, 50, 16};
  const float* cin[5] = {P2, C1, C2, C3, C4};
  float* cout[5] = {C1, C2, C3, C4, C5};
  for (int l = 0; l < 5; ++l) {
    int total = dims[l + 1] * dims[l + 1];
    conv_bn_prelu<<<(total + 255) / 256, 256, 0, stream>>>(
        cin[l], dims[l], cout[l], dims[l + 1], conv_w, conv_b, bn_g, bn_b,
        bn_m, bn_v, prelua, l);
  }

  head_kernel<<<1, 128, 0, stream>>>(C5, fc5_w, fc5_b, ln_g, ln_b, pre5,
                                     fc6_w, fc6_b, out);
}